// GCNEncoder_45346264711627
// MI455X (gfx1250) — compile-verified
//
#include <hip/hip_runtime.h>
#include <hip/hip_bf16.h>

typedef float v2f __attribute__((ext_vector_type(2)));
typedef float v8f __attribute__((ext_vector_type(8)));

// ---- problem constants (fixed by the reference) ----
#define K1 132          // IN_DIM + CONF_EMB
#define H1 64           // HIDDEN
#define H2 32           // OUT
#define CEMB 4
#define INDIM 128

// --------------------------------------------------------------------------
// degree / normalization
// --------------------------------------------------------------------------
__global__ void k_init_deg(float* deg, int N) {
    int i = blockIdx.x * blockDim.x + threadIdx.x;
    if (i < N) deg[i] = 1.0f;                       // self-loop weight
}

__global__ void k_deg_accum(const int* __restrict__ ei, const float* __restrict__ ew,
                            float* deg, int E) {
    int e = blockIdx.x * blockDim.x + threadIdx.x;
    if (e < E) {
        int c = ei[E + e];                          // col
        unsafeAtomicAdd(&deg[c], ew[e]);
    }
}

__global__ void k_deg_to_dinv(float* deg, int N) {
    int i = blockIdx.x * blockDim.x + threadIdx.x;
    if (i < N) {
        float d = deg[i];
        deg[i] = d > 0.0f ? rsqrtf(d) : 0.0f;
    }
}

// --------------------------------------------------------------------------
// h = concat(x, conf_table[conf_ids])   [N, 132]
// --------------------------------------------------------------------------
__global__ void k_build_h(const float* __restrict__ x, const int* __restrict__ conf_ids,
                          const float* __restrict__ conf_table, float* __restrict__ h,
                          int N) {
    int i = blockIdx.x;
    int j = threadIdx.x;                            // blockDim = 160 (5 waves)
    if (i >= N) return;
    if (j < INDIM) {
        h[(size_t)i * K1 + j] = x[(size_t)i * INDIM + j];
    } else if (j < K1) {
        h[(size_t)i * K1 + j] = conf_table[conf_ids[i] * CEMB + (j - INDIM)];
    }
}

// --------------------------------------------------------------------------
// GEMM1: hW1[N,64] = h[N,132] @ W1[132,64]   via V_WMMA_F32_16X16X4_F32
// block = 256 threads = 8 waves: wave w -> rowTile = blk*2 + (w>>2), colTile = w&3
//
// A rows >= N are clamped to a valid address and loaded unconditionally:
// A[m][k] only feeds D[m][*], and rows m >= N are never stored, so no
// per-lane predication is needed (keeps EXEC manipulation out of the loop
// and enables 8B-aligned b64 loads of the two A elements).
// --------------------------------------------------------------------------
__global__ void k_gemm1_wmma(const float* __restrict__ h, const float* __restrict__ W1,
                             float* __restrict__ hW1, int N, int rowTiles) {
    const int lane = threadIdx.x & 31;
    const int wave = threadIdx.x >> 5;
    const int rowTile = blockIdx.x * 2 + (wave >> 2);
    if (rowTile >= rowTiles) return;                // wave-uniform exit
    const int colTile = wave & 3;
    const int m0 = rowTile * 16;
    const int n0 = colTile * 16;
    const int idx  = lane & 15;
    const int half = lane >> 4;
    const int kb   = half * 2;

    int mrow = m0 + idx;
    if (mrow > N - 1) mrow = N - 1;                 // clamp: stay in-bounds
    const float* arow = h + (size_t)mrow * K1 + kb; // 8B-aligned (K1 = 4*33, kb even)
    const float* bcol = W1 + (size_t)kb * H1 + n0 + idx;

    v8f acc = {};
#pragma unroll 4
    for (int k = 0; k < K1; k += 4) {
        v2f a = *(const v2f*)(arow + k);            // global_load_b64
        v2f b;
        b.x = bcol[(size_t)k * H1];
        b.y = bcol[(size_t)k * H1 + H1];
        acc = __builtin_amdgcn_wmma_f32_16x16x4_f32(
            false, a, false, b, (short)0, acc, false, false);
    }

#pragma unroll
    for (int r = 0; r < 8; ++r) {
        int m = m0 + r + half * 8;
        if (m < N) hW1[(size_t)m * H1 + n0 + idx] = acc[r];
    }
}

// --------------------------------------------------------------------------
// GEMM2: hW2[N,32] = h1[N,64] @ W2[64,32]
// block = 256 threads = 8 waves: wave w -> rowTile = blk*4 + (w>>1), colTile = w&1
// --------------------------------------------------------------------------
__global__ void k_gemm2_wmma(const float* __restrict__ h1, const float* __restrict__ W2,
                             float* __restrict__ hW2, int N, int rowTiles) {
    const int lane = threadIdx.x & 31;
    const int wave = threadIdx.x >> 5;
    const int rowTile = blockIdx.x * 4 + (wave >> 1);
    if (rowTile >= rowTiles) return;                // wave-uniform exit
    const int colTile = wave & 1;
    const int m0 = rowTile * 16;
    const int n0 = colTile * 16;
    const int idx  = lane & 15;
    const int half = lane >> 4;
    const int kb   = half * 2;

    int mrow = m0 + idx;
    if (mrow > N - 1) mrow = N - 1;                 // clamp: stay in-bounds
    const float* arow = h1 + (size_t)mrow * H1 + kb; // 8B-aligned (H1 = 64, kb even)
    const float* bcol = W2 + (size_t)kb * H2 + n0 + idx;

    v8f acc = {};
#pragma unroll 4
    for (int k = 0; k < H1; k += 4) {
        v2f a = *(const v2f*)(arow + k);            // global_load_b64
        v2f b;
        b.x = bcol[(size_t)k * H2];
        b.y = bcol[(size_t)k * H2 + H2];
        acc = __builtin_amdgcn_wmma_f32_16x16x4_f32(
            false, a, false, b, (short)0, acc, false, false);
    }

#pragma unroll
    for (int r = 0; r < 8; ++r) {
        int m = m0 + r + half * 8;
        if (m < N) hW2[(size_t)m * H2 + n0 + idx] = acc[r];
    }
}

// --------------------------------------------------------------------------
// layer-1 aggregation
// --------------------------------------------------------------------------
__global__ void k_agg1_init(const float* __restrict__ hW1, const float* __restrict__ dinv,
                            const float* __restrict__ b1, float* __restrict__ o1, int N) {
    int t = blockIdx.x * blockDim.x + threadIdx.x;
    if (t < N * H1) {
        int i = t >> 6;                              // /64
        int j = t & 63;
        float di = dinv[i];
        o1[t] = hW1[t] * di * di + b1[j];            // self-loop (w=1) + bias
    }
}

__global__ void k_agg1_edges(const int* __restrict__ ei, const float* __restrict__ ew,
                             const float* __restrict__ dinv, const float* __restrict__ hW1,
                             float* __restrict__ o1, int E) {
    int e = blockIdx.x * 8 + (threadIdx.x >> 5);     // 8 edges per 256-thread block
    if (e >= E) return;
    int lane = threadIdx.x & 31;
    int r = ei[e];
    int c = ei[E + e];
    float norm = dinv[r] * ew[e] * dinv[c];
    const float* src = hW1 + (size_t)r * H1;
    float*       dst = o1  + (size_t)c * H1;
    unsafeAtomicAdd(&dst[lane],      src[lane]      * norm);
    unsafeAtomicAdd(&dst[lane + 32], src[lane + 32] * norm);
}

__global__ void k_relu_inplace(float* __restrict__ o1, int n) {
    int t = blockIdx.x * blockDim.x + threadIdx.x;
    if (t < n) o1[t] = fmaxf(o1[t], 0.0f);
}

// --------------------------------------------------------------------------
// layer-2 aggregation (into d_out)
// --------------------------------------------------------------------------
__global__ void k_agg2_init(const float* __restrict__ hW2, const float* __restrict__ dinv,
                            const float* __restrict__ b2, float* __restrict__ out, int N) {
    int t = blockIdx.x * blockDim.x + threadIdx.x;
    if (t < N * H2) {
        int i = t >> 5;                              // /32
        int j = t & 31;
        float di = dinv[i];
        out[t] = hW2[t] * di * di + b2[j];
    }
}

__global__ void k_agg2_edges(const int* __restrict__ ei, const float* __restrict__ ew,
                             const float* __restrict__ dinv, const float* __restrict__ hW2,
                             float* __restrict__ out, int E) {
    int e = blockIdx.x * 8 + (threadIdx.x >> 5);
    if (e >= E) return;
    int lane = threadIdx.x & 31;                     // 32 features == wave32
    int r = ei[e];
    int c = ei[E + e];
    float norm = dinv[r] * ew[e] * dinv[c];
    unsafeAtomicAdd(&out[(size_t)c * H2 + lane], hW2[(size_t)r * H2 + lane] * norm);
}

// --------------------------------------------------------------------------
extern "C" void kernel_launch(void* const* d_in, const int* in_sizes, int n_in,
                              void* d_out, int out_size, void* d_ws, size_t ws_size,
                              hipStream_t stream) {
    const float* x          = (const float*)d_in[0];
    const int*   conf_ids   = (const int*)  d_in[1];
    const int*   edge_index = (const int*)  d_in[2];
    const float* edge_weight= (const float*)d_in[3];
    const float* conf_table = (const float*)d_in[4];
    const float* W1         = (const float*)d_in[5];
    const float* b1         = (const float*)d_in[6];
    const float* W2         = (const float*)d_in[7];
    const float* b2         = (const float*)d_in[8];
    float* out = (float*)d_out;

    const int N = in_sizes[1];          // 50000
    const int E = in_sizes[3];          // 1600000

    // workspace carve-out (floats): dinv | h | hW1 | o1 | hW2   (~59 MB)
    float* ws   = (float*)d_ws;
    float* dinv = ws;
    float* h    = dinv + N;
    float* hW1  = h    + (size_t)N * K1;
    float* o1   = hW1  + (size_t)N * H1;
    float* hW2  = o1   + (size_t)N * H1;

    const int rowTiles = (N + 15) / 16;             // 3125

    // 1) degree -> dinv
    k_init_deg   <<<(N + 255) / 256, 256, 0, stream>>>(dinv, N);
    k_deg_accum  <<<(E + 255) / 256, 256, 0, stream>>>(edge_index, edge_weight, dinv, E);
    k_deg_to_dinv<<<(N + 255) / 256, 256, 0, stream>>>(dinv, N);

    // 2) h = [x | emb]
    k_build_h<<<N, 160, 0, stream>>>(x, conf_ids, conf_table, h, N);

    // 3) hW1 = h @ W1  (WMMA f32)
    k_gemm1_wmma<<<(rowTiles + 1) / 2, 256, 0, stream>>>(h, W1, hW1, N, rowTiles);

    // 4) layer-1 aggregate + bias + relu
    k_agg1_init <<<(N * H1 + 255) / 256, 256, 0, stream>>>(hW1, dinv, b1, o1, N);
    k_agg1_edges<<<(E + 7) / 8, 256, 0, stream>>>(edge_index, edge_weight, dinv, hW1, o1, E);
    k_relu_inplace<<<(N * H1 + 255) / 256, 256, 0, stream>>>(o1, N * H1);

    // 5) hW2 = relu(o1) @ W2  (WMMA f32)
    k_gemm2_wmma<<<(rowTiles + 3) / 4, 256, 0, stream>>>(o1, W2, hW2, N, rowTiles);

    // 6) layer-2 aggregate into d_out
    k_agg2_init <<<(N * H2 + 255) / 256, 256, 0, stream>>>(hW2, dinv, b2, out, N);
    k_agg2_edges<<<(E + 7) / 8, 256, 0, stream>>>(edge_index, edge_weight, dinv, hW2, out, E);
}